// LiveNet_22256520528694
// MI455X (gfx1250) — compile-verified
//
#include <hip/hip_runtime.h>

// MI455X / gfx1250: fp32 MLP  y = relu(x@W1 + b1) @ W2 + b2
// Memory-bound (296 MB @ 23.3 TB/s ~ 12.7 us floor); all matmul math via
// V_WMMA_F32_16X16X4_F32 (exact fp32). Weights held in VGPRs across 4 tiles
// per wave to amortize the B-matrix preload; x streamed with b64 A-layout
// loads + global_prefetch for the next tile.

typedef __attribute__((ext_vector_type(2))) float v2f;
typedef __attribute__((ext_vector_type(8))) float v8f;

#define BATCH 1048576
#define N_IN  64
#define N_MID 32
#define N_OUT 10

#define ROWS_PER_WAVE   16
#define TILES_PER_WAVE  4
#define WAVES_PER_BLOCK 8
#define ROWS_PER_BLOCK  (ROWS_PER_WAVE * TILES_PER_WAVE * WAVES_PER_BLOCK)  // 512
#define LDS_STRIDE      36   // 32 cols + 4 pad dwords -> conflict-free b64 reads

__global__ __launch_bounds__(256)
void livenet_mlp_wmma(const float* __restrict__ x,
                      const float* __restrict__ W1,
                      const float* __restrict__ b1,
                      const float* __restrict__ W2,
                      const float* __restrict__ b2,
                      float* __restrict__ y)
{
    // per-wave 16x32 h tile staging (layer1 D-layout -> layer2 A-layout transpose)
    __shared__ float lds_h[WAVES_PER_BLOCK * ROWS_PER_WAVE * LDS_STRIDE];

    const int tid  = threadIdx.x;
    const int lane = tid & 31;
    const int wave = tid >> 5;
    const int j    = lane & 15;   // column/row-within-half index
    const int hi   = lane >> 4;   // 0: lanes 0-15, 1: lanes 16-31

    // wave covers rows [wrow0, wrow0 + 64)
    const long wrow0 = ((long)blockIdx.x * WAVES_PER_BLOCK + wave) *
                       (ROWS_PER_WAVE * TILES_PER_WAVE);

    float* hbuf = lds_h + wave * (ROWS_PER_WAVE * LDS_STRIDE);

    // ---- preload W1 in B-matrix (4x16, KxN) layout: 2 N-tiles x 16 K-steps ----
    // B tile (n,k) lane(j,hi): b.x = W1[4k+2hi][16n+j], b.y = W1[4k+1+2hi][16n+j]
    v2f B1[2][16];
    #pragma unroll
    for (int n = 0; n < 2; ++n) {
        #pragma unroll
        for (int k = 0; k < 16; ++k) {
            const int r = 4 * k + 2 * hi;
            const int c = 16 * n + j;
            v2f b;
            b.x = W1[(long)r * N_MID + c];
            b.y = W1[(long)(r + 1) * N_MID + c];
            B1[n][k] = b;
        }
    }

    // ---- preload W2 in B-layout, N padded 10 -> 16 with zeros ----
    v2f B2[8];
    #pragma unroll
    for (int k = 0; k < 8; ++k) {
        const int r = 4 * k + 2 * hi;
        v2f b;
        b.x = (j < N_OUT) ? W2[(long)r * N_OUT + j]       : 0.0f;
        b.y = (j < N_OUT) ? W2[(long)(r + 1) * N_OUT + j] : 0.0f;
        B2[k] = b;
    }

    const float b1v0 = b1[j];
    const float b1v1 = b1[16 + j];
    const float b2v  = (j < N_OUT) ? b2[j] : 0.0f;

    #pragma unroll
    for (int t = 0; t < TILES_PER_WAVE; ++t) {
        const long row0 = wrow0 + (long)t * ROWS_PER_WAVE;
        const float* xrow = x + (row0 + j) * (long)N_IN;

        // prefetch next tile's x rows into cache (global_prefetch_b8)
        if (t + 1 < TILES_PER_WAVE) {
            __builtin_prefetch(xrow + (long)ROWS_PER_WAVE * N_IN, 0, 3);
        }

        // ---- layer 1: h[16x32] = relu(x[16x64] @ W1 + b1), 32 fp32 WMMAs ----
        // A (16x4 fp32): lane j holds row j {K, K+1}; lane 16+j holds {K+2, K+3}
        v8f acc0 = {};
        v8f acc1 = {};
        #pragma unroll
        for (int k = 0; k < 16; ++k) {
            const v2f a = *(const v2f*)(xrow + 4 * k + 2 * hi);  // global_load_b64
            acc0 = __builtin_amdgcn_wmma_f32_16x16x4_f32(
                false, a, false, B1[0][k], (short)0, acc0, false, false);
            acc1 = __builtin_amdgcn_wmma_f32_16x16x4_f32(
                false, a, false, B1[1][k], (short)0, acc1, false, false);
        }

        // bias + ReLU in C/D layout (col uniform per lane), spill h to LDS
        #pragma unroll
        for (int v = 0; v < 8; ++v) {
            const int m = v + 8 * hi;                      // output row of this VGPR
            hbuf[m * LDS_STRIDE + j]      = fmaxf(acc0[v] + b1v0, 0.0f);
            hbuf[m * LDS_STRIDE + 16 + j] = fmaxf(acc1[v] + b1v1, 0.0f);
        }

        // ---- layer 2: y[16x16(pad)] = h[16x32] @ W2, 8 fp32 WMMAs ----
        // A read back from LDS in A-layout (same-wave LDS is in-order, no barrier)
        v8f accy = {};
        #pragma unroll
        for (int k = 0; k < 8; ++k) {
            const v2f a2 = *(const v2f*)(hbuf + j * LDS_STRIDE + 4 * k + 2 * hi); // ds_load_b64
            accy = __builtin_amdgcn_wmma_f32_16x16x4_f32(
                false, a2, false, B2[k], (short)0, accy, false, false);
        }

        // ---- bias + store the 10 valid output columns ----
        if (j < N_OUT) {
            #pragma unroll
            for (int v = 0; v < 8; ++v) {
                const long m = row0 + v + 8 * hi;
                y[m * N_OUT + j] = accy[v] + b2v;
            }
        }
    }
}

extern "C" void kernel_launch(void* const* d_in, const int* in_sizes, int n_in,
                              void* d_out, int out_size, void* d_ws, size_t ws_size,
                              hipStream_t stream) {
    (void)in_sizes; (void)n_in; (void)out_size; (void)d_ws; (void)ws_size;
    const float* x  = (const float*)d_in[0];
    const float* W1 = (const float*)d_in[1];
    const float* b1 = (const float*)d_in[2];
    const float* W2 = (const float*)d_in[3];
    const float* b2 = (const float*)d_in[4];
    float* y = (float*)d_out;

    dim3 grid(BATCH / ROWS_PER_BLOCK);   // 2048 blocks
    dim3 block(256);                     // 8 waves (wave32)
    livenet_mlp_wmma<<<grid, block, 0, stream>>>(x, W1, b1, W2, b2, y);
}